// SNIPER_19112604467876
// MI455X (gfx1250) — compile-verified
//
#include <hip/hip_runtime.h>

// Problem constants (from reference): T=8, L=8, N=500, D=64, K=12, S=2
#define T_    8
#define L_    8
#define N_    500
#define D_    64
#define K_    12
#define LN_   (L_*N_)        // 4000
#define LND_  (LN_*D_)       // 256000
#define TLN_  (T_*LN_)       // 32000
#define TLND_ (TLN_*D_)      // 2048000

typedef __attribute__((ext_vector_type(2))) float v2f;
typedef __attribute__((ext_vector_type(4))) float v4f;
typedef __attribute__((ext_vector_type(8))) float v8f;

__device__ __forceinline__ float sigf(float x) { return 1.0f / (1.0f + __expf(-x)); }

// Pair-interleaved LDS layout for a 64x64 fp32 weight matrix:
//   sW[((p*16 + lcol)*4 + nt)*2 + w] = W[2p + w][nt*16 + lcol]
// so one lane's 4 n-tile B fragments for k-rows {2p, 2p+1} are 32 contiguous
// bytes -> two ds_load_b128 per k-step, operands land in even-aligned pairs.
__device__ __forceinline__ int pairIdx(int row, int col) {
  return (((row >> 1) * 16 + (col & 15)) * 4 + (col >> 4)) * 2 + (row & 1);
}

#define WMMA_F32(A, B, C) \
  __builtin_amdgcn_wmma_f32_16x16x4_f32(false, (A), false, (B), (short)0, (C), false, false)

// ---------------------------------------------------------------------------
// Fused dual-projection WMMA GEMM:  Q = X@Wq,  Kf = X@Wk   (rows = 32000)
// One wave owns a 16-row strip; each A fragment feeds 8 WMMAs.
// ---------------------------------------------------------------------------
__global__ __launch_bounds__(256) void gemm64_qk(const float* __restrict__ X,
                                                 const float* __restrict__ Wq,
                                                 const float* __restrict__ Wk,
                                                 float* __restrict__ Q,
                                                 float* __restrict__ Kf) {
  __shared__ __align__(16) float sW[2][4096];
  for (int i = threadIdx.x; i < 4096; i += 256) {
    const int dst = pairIdx(i >> 6, i & 63);
    sW[0][dst] = Wq[i];
    sW[1][dst] = Wk[i];
  }
  __syncthreads();

  const int wave  = threadIdx.x >> 5;
  const int lane  = threadIdx.x & 31;
  const int tile  = blockIdx.x * 8 + wave;  // 2000 tiles over 250 blocks
  const int row0  = tile << 4;
  const int halfk = (lane >> 4) << 1;       // lanes 0-15 -> K {0,1}; 16-31 -> K {2,3}
  const int lcol  = lane & 15;

  const float* xp = X + (size_t)(row0 + lcol) * 64 + halfk;

  v8f q0 = {}, q1 = {}, q2 = {}, q3 = {};
  v8f c0 = {}, c1 = {}, c2 = {}, c3 = {};
#pragma unroll
  for (int kk = 0; kk < 16; ++kk) {
    v2f a;
    a.x = xp[kk * 4];
    a.y = xp[kk * 4 + 1];
    const int fb = (((kk * 2 + (halfk >> 1)) * 16) + lcol) * 8;
    {
      const v4f lo = *(const v4f*)&sW[0][fb];
      const v4f hi = *(const v4f*)&sW[0][fb + 4];
      v2f b0 = {lo.x, lo.y}, b1 = {lo.z, lo.w}, b2 = {hi.x, hi.y}, b3 = {hi.z, hi.w};
      q0 = WMMA_F32(a, b0, q0);
      q1 = WMMA_F32(a, b1, q1);
      q2 = WMMA_F32(a, b2, q2);
      q3 = WMMA_F32(a, b3, q3);
    }
    {
      const v4f lo = *(const v4f*)&sW[1][fb];
      const v4f hi = *(const v4f*)&sW[1][fb + 4];
      v2f b0 = {lo.x, lo.y}, b1 = {lo.z, lo.w}, b2 = {hi.x, hi.y}, b3 = {hi.z, hi.w};
      c0 = WMMA_F32(a, b0, c0);
      c1 = WMMA_F32(a, b1, c1);
      c2 = WMMA_F32(a, b2, c2);
      c3 = WMMA_F32(a, b3, c3);
    }
  }

  // C/D layout: VGPR v -> row = row0 + v + (lane>=16 ? 8 : 0), col = nt*16 + lcol
  const int rbase = row0 + ((lane >> 4) << 3);
#pragma unroll
  for (int v = 0; v < 8; ++v) {
    const size_t ro = (size_t)(rbase + v) * 64 + lcol;
    Q[ro]      = q0[v];
    Q[ro + 16] = q1[v];
    Q[ro + 32] = q2[v];
    Q[ro + 48] = q3[v];
    Kf[ro]      = c0[v];
    Kf[ro + 16] = c1[v];
    Kf[ro + 32] = c2[v];
    Kf[ro + 48] = c3[v];
  }
}

// ---------------------------------------------------------------------------
// Single-projection WMMA GEMM with fused epilogue (wd stage):
// MODE 1: Y = sigmoid(X@W) ; MODE 2: acc += sigmoid(X@W) * mix[col]
// ---------------------------------------------------------------------------
template <int MODE>
__global__ __launch_bounds__(256) void gemm64_wmma(const float* __restrict__ X,
                                                   const float* __restrict__ W,
                                                   float* __restrict__ Y,
                                                   const float* __restrict__ mixvec,
                                                   float* __restrict__ acc) {
  __shared__ __align__(16) float sW[4096];
  for (int i = threadIdx.x; i < 4096; i += 256) sW[pairIdx(i >> 6, i & 63)] = W[i];
  __syncthreads();

  const int wave  = threadIdx.x >> 5;
  const int lane  = threadIdx.x & 31;
  const int tile  = blockIdx.x * 8 + wave;
  const int row0  = tile << 4;
  const int halfk = (lane >> 4) << 1;
  const int lcol  = lane & 15;

  const float* xp = X + (size_t)(row0 + lcol) * 64 + halfk;

  v8f c0 = {}, c1 = {}, c2 = {}, c3 = {};
#pragma unroll
  for (int kk = 0; kk < 16; ++kk) {
    v2f a;
    a.x = xp[kk * 4];
    a.y = xp[kk * 4 + 1];
    const int fb = (((kk * 2 + (halfk >> 1)) * 16) + lcol) * 8;
    const v4f lo = *(const v4f*)&sW[fb];
    const v4f hi = *(const v4f*)&sW[fb + 4];
    v2f b0 = {lo.x, lo.y}, b1 = {lo.z, lo.w}, b2 = {hi.x, hi.y}, b3 = {hi.z, hi.w};
    c0 = WMMA_F32(a, b0, c0);
    c1 = WMMA_F32(a, b1, c1);
    c2 = WMMA_F32(a, b2, c2);
    c3 = WMMA_F32(a, b3, c3);
  }

  const int rbase = row0 + ((lane >> 4) << 3);
#pragma unroll
  for (int v = 0; v < 8; ++v) {
    const int r = rbase + v;
    float vals[4] = {c0[v], c1[v], c2[v], c3[v]};
#pragma unroll
    for (int nt = 0; nt < 4; ++nt) {
      const int col = nt * 16 + lcol;
      if (MODE == 1) {
        Y[(size_t)r * 64 + col] = sigf(vals[nt]);
      } else {
        acc[(size_t)r * 64 + col] += sigf(vals[nt]) * mixvec[col];
      }
    }
  }
}

// ---------------------------------------------------------------------------
// evo scan step: new = sigmoid((cat[now,st] @ w1) * th + now*(1-th))
// ---------------------------------------------------------------------------
__global__ void evo_step_kernel(const float* __restrict__ now, const float* __restrict__ st,
                                const float* __restrict__ th, const float* __restrict__ w1,
                                float* __restrict__ dyn_out, float* __restrict__ diff_out,
                                float* __restrict__ now_final) {
  const int tid = blockIdx.x * blockDim.x + threadIdx.x;
  if (tid >= LN_ * D_) return;
  const int r = tid >> 6;
  const int e = tid & 63;
  const float* nr = now + (size_t)r * 64;
  const float* sr = st + (size_t)r * 64;
  float g = 0.0f;
#pragma unroll 8
  for (int d = 0; d < 64; ++d) {
    g += nr[d] * w1[d * 64 + e];
    g += sr[d] * w1[(64 + d) * 64 + e];
  }
  const float t = th[r];
  const float prev = nr[e];
  const float val = sigf(g * t + prev * (1.0f - t));
  dyn_out[tid] = val;
  diff_out[tid] = val - prev;
  if (now_final) now_final[tid] = val;
}

// ---------------------------------------------------------------------------
// Attention core on pre-projected Q/Kf (gather commutes with @wk):
//   logits[j] = <q_r, Kf[neigh_j]> ; a = softmax_j ; Out = q + sum_j a_j*k_j
// One wave32 per row; lane handles elements {lane, lane+32}. In-place Out==Q.
// ---------------------------------------------------------------------------
__global__ __launch_bounds__(256) void attn_kernel(const float* __restrict__ Q,
                                                   const float* __restrict__ Kf,
                                                   const int* __restrict__ neigh,
                                                   float* __restrict__ Out) {
  const int r = blockIdx.x * 8 + (threadIdx.x >> 5);
  const int lane = threadIdx.x & 31;
  if (r >= TLN_) return;  // wave-uniform
  const int n = r % N_;
  const int base = r - n;

  const float* qr = Q + (size_t)r * 64;
  const float q0 = qr[lane];
  const float q1 = qr[lane + 32];

  float k0[K_], k1[K_], lg[K_];
#pragma unroll
  for (int j = 0; j < K_; ++j) {
    const int m = neigh[n * K_ + j];
    const float* kr = Kf + (size_t)(base + m) * 64;
    k0[j] = kr[lane];
    k1[j] = kr[lane + 32];
    float p = q0 * k0[j] + q1 * k1[j];
#pragma unroll
    for (int off = 16; off >= 1; off >>= 1) p += __shfl_xor(p, off, 32);
    lg[j] = p;
  }
  float mx = lg[0];
#pragma unroll
  for (int j = 1; j < K_; ++j) mx = fmaxf(mx, lg[j]);
  float s = 0.0f;
#pragma unroll
  for (int j = 0; j < K_; ++j) {
    lg[j] = __expf(lg[j] - mx);
    s += lg[j];
  }
  const float inv = 1.0f / s;
  float o0 = q0, o1 = q1;
#pragma unroll
  for (int j = 0; j < K_; ++j) {
    const float a = lg[j] * inv;
    o0 += a * k0[j];
    o1 += a * k1[j];
  }
  Out[(size_t)r * 64 + lane] = o0;
  Out[(size_t)r * 64 + 32 + lane] = o1;
}

__global__ void zero_kernel(float* __restrict__ p, int n) {
  const int i = blockIdx.x * blockDim.x + threadIdx.x;
  if (i < n) p[i] = 0.0f;
}

__global__ void sigmoid_kernel(const float* __restrict__ in, float* __restrict__ out, int n) {
  const int i = blockIdx.x * blockDim.x + threadIdx.x;
  if (i < n) out[i] = sigf(in[i]);
}

// ---------------------------------------------------------------------------
// conv-LSTM scan over L with scalar h/c per (t,n).
// ---------------------------------------------------------------------------
__global__ void lstm_kernel(const float* __restrict__ dyn, const float* __restrict__ sta,
                            const float* __restrict__ cw, const float* __restrict__ cb,
                            float* __restrict__ hlast) {
  const int idx = blockIdx.x * blockDim.x + threadIdx.x;
  if (idx >= T_ * N_) return;
  const int t = idx / N_;
  const int n = idx % N_;
  float h = 0.0f, c = 0.0f;
  for (int l = 0; l < L_; ++l) {
    const size_t ro = ((size_t)(t * L_ + l) * N_ + n) * 64;
    float gs[4];
#pragma unroll
    for (int gi = 0; gi < 4; ++gi) gs[gi] = cb[gi] + h * cw[gi * 129 + 128];
    for (int d = 0; d < 64; ++d) {
      const float xd = dyn[ro + d];
      const float xs = sta[ro + d];
#pragma unroll
      for (int gi = 0; gi < 4; ++gi) gs[gi] += xd * cw[gi * 129 + d] + xs * cw[gi * 129 + 64 + d];
    }
    const float ig = sigf(gs[0]);
    const float fg = sigf(gs[1]);
    const float og = sigf(gs[2]);
    const float gg = tanhf(gs[3]);
    c = fg * c + ig * gg;
    h = og * tanhf(c);
  }
  hlast[idx] = h;
}

// out[t,i] = sigmoid(sum_n h[t,n]*fin_w[i,n] + fin_b[i])
__global__ void final_kernel(const float* __restrict__ hl, const float* __restrict__ fw,
                             const float* __restrict__ fb, float* __restrict__ out) {
  const int idx = blockIdx.x * blockDim.x + threadIdx.x;
  if (idx >= T_ * N_) return;
  const int t = idx / N_;
  const int i = idx % N_;
  float s = fb[i];
  const float* hr = hl + (size_t)t * N_;
  const float* wr = fw + (size_t)i * N_;
  for (int n = 0; n < N_; ++n) s += hr[n] * wr[n];
  out[idx] = sigf(s);
}

// ---------------------------------------------------------------------------
extern "C" void kernel_launch(void* const* d_in, const int* in_sizes, int n_in,
                              void* d_out, int out_size, void* d_ws, size_t ws_size,
                              hipStream_t stream) {
  (void)in_sizes; (void)n_in; (void)out_size; (void)ws_size;

  const float* all_data_static      = (const float*)d_in[0];   // (T,L,N,D)
  const float* thre_nc              = (const float*)d_in[1];   // (T,L,N,1)
  const float* all_data_dynamic_now = (const float*)d_in[2];   // (L,N,D)
  const int*   neigh_poi            = (const int*)d_in[3];     // (N,K)
  const int*   neigh_road           = (const int*)d_in[4];
  const int*   neigh_record         = (const int*)d_in[5];
  const float* w1                   = (const float*)d_in[6];   // (2D,D)
  const float* wq_all               = (const float*)d_in[7];   // (2,3,S,D,D)
  const float* wk_all               = (const float*)d_in[8];
  const float* wd_all               = (const float*)d_in[9];
  const float* wmix_all             = (const float*)d_in[10];  // (2,3,D)
  const float* conv_w               = (const float*)d_in[11];  // (4,129)
  const float* conv_b               = (const float*)d_in[12];  // (4,)
  const float* fin_w                = (const float*)d_in[13];  // (N,N)
  const float* fin_b                = (const float*)d_in[14];  // (N,)

  // Output layout: out (T,N)=4000 | now_final (L,N,D)=256000 | diffs (7,L,N,D)
  float* out_main = (float*)d_out;
  float* nf_out   = out_main + T_ * N_;
  float* diffs    = nf_out + LND_;

  // Workspace layout (floats)
  float* ws      = (float*)d_ws;
  float* all_dyn = ws;                 // (T,L,N,D)
  float* Qb      = ws + 1ull * TLND_;  // Q / attention-out (in place)
  float* Kb      = ws + 2ull * TLND_;  // per-node projected keys
  float* Bb      = ws + 3ull * TLND_;  // stage-0 intermediate
  float* accb    = ws + 4ull * TLND_;  // wmix accumulator
  float* dynres  = ws + 5ull * TLND_;
  float* stares  = ws + 6ull * TLND_;
  float* hlast   = ws + 7ull * TLND_;  // (T,N)

  // ---- Phase A: evo scan (7 sequential GEMM+elementwise steps) ----
  hipMemcpyAsync(all_dyn, all_data_dynamic_now, sizeof(float) * LND_,
                 hipMemcpyDeviceToDevice, stream);
  for (int t = 1; t < T_; ++t) {
    evo_step_kernel<<<(LN_ * D_ + 255) / 256, 256, 0, stream>>>(
        all_dyn + (size_t)(t - 1) * LND_, all_data_static + (size_t)t * LND_,
        thre_nc + (size_t)t * LN_, w1, all_dyn + (size_t)t * LND_,
        diffs + (size_t)(t - 1) * LND_, (t == T_ - 1) ? nf_out : nullptr);
  }

  // ---- Phase B: two multiattentions (dyn, static) ----
  const int* neighs[3] = {neigh_poi, neigh_road, neigh_record};
  const int GEMM_BLOCKS = TLN_ / 16 / 8;  // 250
  const int EW_BLOCKS = (TLND_ + 255) / 256;

  auto multiatt = [&](int m, const float* src, float* res) {
    zero_kernel<<<EW_BLOCKS, 256, 0, stream>>>(accb, TLND_);
    for (int g = 0; g < 3; ++g) {
      const float* X = src;
      for (int s = 0; s < 2; ++s) {
        const size_t woff = (size_t)(((m * 3 + g) * 2) + s) * 4096;
        gemm64_qk<<<GEMM_BLOCKS, 256, 0, stream>>>(X, wq_all + woff, wk_all + woff, Qb, Kb);
        attn_kernel<<<TLN_ / 8, 256, 0, stream>>>(Qb, Kb, neighs[g], Qb);
        if (s == 0) {
          gemm64_wmma<1><<<GEMM_BLOCKS, 256, 0, stream>>>(Qb, wd_all + woff, Bb, nullptr, nullptr);
          X = Bb;
        } else {
          gemm64_wmma<2><<<GEMM_BLOCKS, 256, 0, stream>>>(
              Qb, wd_all + woff, nullptr, wmix_all + (size_t)(m * 3 + g) * 64, accb);
        }
      }
    }
    sigmoid_kernel<<<EW_BLOCKS, 256, 0, stream>>>(accb, res, TLND_);
  };
  multiatt(0, all_dyn, dynres);
  multiatt(1, all_data_static, stares);

  // ---- Phase C: LSTM scan over L ----
  lstm_kernel<<<(T_ * N_ + 255) / 256, 256, 0, stream>>>(dynres, stares, conv_w, conv_b, hlast);

  // ---- Phase D: final projection + sigmoid ----
  final_kernel<<<(T_ * N_ + 255) / 256, 256, 0, stream>>>(hlast, fin_w, fin_b, out_main);
}